// ConfigurableRGCN_3375844295101
// MI455X (gfx1250) — compile-verified
//
#include <hip/hip_runtime.h>

// Problem constants (match reference)
#define NNODES 20000
#define NEDGES 320000
#define RREL   40
#define NBASES 34
#define DDIM   128
#define KREL   (RREL * DDIM)        // 5120
#define KTOT   ((RREL + 1) * DDIM)  // 5248 (root folded in as extra relation)
#define KSTEPS (KTOT / 4)           // 1312 WMMA K-steps (K=4 each)

typedef float v2f __attribute__((ext_vector_type(2)));
typedef float v8f __attribute__((ext_vector_type(8)));

// ---------------------------------------------------------------------------
// x_gathered[n,:] = node_emb[node_index[n],:]   (32 lanes/node, float4 each)
// ---------------------------------------------------------------------------
__global__ void rgcn_gather_kernel(const int* __restrict__ node_index,
                                   const float* __restrict__ node_emb,
                                   float* __restrict__ xg) {
    int t = blockIdx.x * blockDim.x + threadIdx.x;
    int n = t >> 5;
    int c = (t & 31) << 2;
    if (n < NNODES) {
        long src = (long)node_index[n] * DDIM + c;
        *(float4*)(xg + (long)n * DDIM + c) = *(const float4*)(node_emb + src);
    }
}

// ---------------------------------------------------------------------------
// Edge scatter: S[dst*R+et, :] += x[src, :];  cnt[dst*R+et] += 1
// One wave (32 lanes) per edge; each lane handles 4 contiguous floats.
// ---------------------------------------------------------------------------
__global__ void rgcn_scatter_kernel(const int* __restrict__ edge_index,
                                    const int* __restrict__ edge_type,
                                    const float* __restrict__ x,
                                    float* __restrict__ S,
                                    float* __restrict__ cnt) {
    int t = blockIdx.x * blockDim.x + threadIdx.x;
    int e = t >> 5;
    int lane = t & 31;
    if (e >= NEDGES) return;
    int src = edge_index[e];
    int dst = edge_index[NEDGES + e];
    int r   = edge_type[e];
    long seg = (long)dst * RREL + r;
    if (lane == 0) atomicAdd(cnt + seg, 1.0f);
    float4 v = *(const float4*)(x + (long)src * DDIM + (lane << 2));
    float* p = S + seg * DDIM + (lane << 2);
    atomicAdd(p + 0, v.x);
    atomicAdd(p + 1, v.y);
    atomicAdd(p + 2, v.z);
    atomicAdd(p + 3, v.w);
}

// ---------------------------------------------------------------------------
// In-place: cnt[i] <- 1 / max(cnt[i], 1)   (exact IEEE divide, done once here
// so the GEMM hot loop only loads the precomputed reciprocal)
// ---------------------------------------------------------------------------
__global__ void rgcn_invcnt_kernel(float* __restrict__ cnt) {
    int i = blockIdx.x * blockDim.x + threadIdx.x;
    if (i < NNODES * RREL) cnt[i] = 1.0f / fmaxf(cnt[i], 1.0f);
}

// ---------------------------------------------------------------------------
// Build Wcat in PACKED WMMA-fragment order.
// Logical Wcat[k,o], k = r*128+i:  sum_b comp[r,b]*bases[b,i,o]  (r<R)
//                                  root[i,o]                     (r==R)
// Packed float2 element u = (ks*8 + t)*32 + lane holds
//   ( Wcat[ks*4 + 2*hi    , t*16+m],
//     Wcat[ks*4 + 2*hi + 1, t*16+m] )     with m=lane&15, hi=lane>>4,
// i.e. exactly the B fragment lane `lane` feeds to v_wmma_f32_16x16x4_f32
// for K-step ks, N-tile t.  GEMM B loads become coalesced global_load_b64.
// ---------------------------------------------------------------------------
__global__ void rgcn_wcat_kernel(const float* __restrict__ comp,
                                 const float* __restrict__ bases,
                                 const float* __restrict__ root,
                                 float2* __restrict__ Wpack) {
    int u = blockIdx.x * blockDim.x + threadIdx.x;
    if (u >= KSTEPS * 8 * 32) return;
    int lane = u & 31;
    int t    = (u >> 5) & 7;
    int ks   = u >> 8;
    int m  = lane & 15;
    int hi = lane >> 4;
    int k  = ks * 4 + 2 * hi;      // even, so k and k+1 share relation r
    int o  = t * 16 + m;
    int r  = k >> 7;
    int i  = k & (DDIM - 1);
    float w0, w1;
    if (r < RREL) {
        w0 = 0.0f; w1 = 0.0f;
        #pragma unroll 2
        for (int b = 0; b < NBASES; ++b) {
            float cb = comp[r * NBASES + b];
            const float* bb = bases + ((long)b * DDIM + i) * DDIM + o;
            w0 += cb * bb[0];
            w1 += cb * bb[DDIM];
        }
    } else {
        w0 = root[i * DDIM + o];
        w1 = root[(i + 1) * DDIM + o];
    }
    Wpack[u] = make_float2(w0, w1);
}

// ---------------------------------------------------------------------------
// WMMA GEMM:  out[n,o] = sum_k A[n,k] * Wcat[k,o] + bias[o]
//   A[n, r*128+i] = S[n,r,i] * inv[n,r]  (r<R);   A[n, R*128+i] = x[n,i]
// Block = 128 threads = 4 waves = 2 row-tiles x 2 K-halves:
//   wave w: rowtile = w&1 (16 rows), khalf = w>>1
//   khalf 0: relations 0..19;  khalf 1: relations 20..39 + root ("relation 40")
// Relation-major nested loop: scale loaded once per relation, inner loop of 32
// K-steps is branch-free: 8 coalesced global_load_b64 + 8 WMMA per step.
// Partials combined through LDS. 625 blocks (20000 = 625*32, no tail).
// ---------------------------------------------------------------------------
__global__ __launch_bounds__(128)
void rgcn_gemm_kernel(const float* __restrict__ S,
                      const float* __restrict__ inv,   // precomputed 1/max(cnt,1)
                      const float* __restrict__ x,
                      const float2* __restrict__ Wp,
                      const float* __restrict__ bias,
                      float* __restrict__ out) {
    __shared__ float lds[2 * 16 * DDIM];   // 16 KB: partial C for 2 row-tiles

    const int wave = threadIdx.x >> 5;
    const int lane = threadIdx.x & 31;
    const int m    = lane & 15;
    const int hi   = lane >> 4;
    const int rowtile = wave & 1;
    const int khalf   = wave >> 1;

    const int row0 = blockIdx.x * 32 + rowtile * 16;
    const int rowm = row0 + m;
    const float* Srow = S + (long)rowm * KREL;
    const float* xrow = x + (long)rowm * DDIM;
    const float* irow = inv + (long)rowm * RREL;

    v8f acc[8] = {};

    const int r_begin = khalf ? (RREL / 2) : 0;
    const int r_end   = khalf ? (RREL + 1) : (RREL / 2);  // khalf1 takes root too

    for (int r = r_begin; r < r_end; ++r) {
        const bool  isrel = (r < RREL);
        const float cinv  = isrel ? irow[r] : 1.0f;
        const float* ap   = (isrel ? (Srow + (long)r * DDIM) : xrow) + 2 * hi;
        const float2* bp  = Wp + (size_t)r * 32 * 256 + lane;
        #pragma unroll 4
        for (int j = 0; j < 32; ++j) {
            float2 av = *(const float2*)(ap + j * 4);
            v2f a; a[0] = av.x * cinv; a[1] = av.y * cinv;
            const float2* bj = bp + (size_t)j * 256;
            #pragma unroll
            for (int t = 0; t < 8; ++t) {
                float2 bv = bj[t * 32];       // coalesced global_load_b64
                v2f b; b[0] = bv.x; b[1] = bv.y;
                acc[t] = __builtin_amdgcn_wmma_f32_16x16x4_f32(
                    false, a, false, b, (short)0, acc[t], false, false);
            }
        }
    }

    if (khalf) {
        // stash partial C tile in LDS
        #pragma unroll
        for (int t = 0; t < 8; ++t) {
            int col = t * 16 + m;
            #pragma unroll
            for (int v = 0; v < 8; ++v)
                lds[rowtile * (16 * DDIM) + (v + 8 * hi) * DDIM + col] = acc[t][v];
        }
    }
    __syncthreads();
    // ---- khalf 0 combines both halves, adds bias, stores ----
    if (!khalf) {
        #pragma unroll
        for (int t = 0; t < 8; ++t) {
            int col = t * 16 + m;
            float bv = bias[col];
            #pragma unroll
            for (int v = 0; v < 8; ++v) {
                int rr = row0 + v + 8 * hi;
                float other = lds[rowtile * (16 * DDIM) + (v + 8 * hi) * DDIM + col];
                out[(long)rr * DDIM + col] = acc[t][v] + other + bv;
            }
        }
    }
}

// ---------------------------------------------------------------------------
extern "C" void kernel_launch(void* const* d_in, const int* in_sizes, int n_in,
                              void* d_out, int out_size, void* d_ws, size_t ws_size,
                              hipStream_t stream) {
    (void)in_sizes; (void)n_in; (void)out_size; (void)ws_size;

    const int*   node_index = (const int*)d_in[0];
    const int*   edge_index = (const int*)d_in[1];
    const int*   edge_type  = (const int*)d_in[2];
    // d_in[3] = node_frequency (unused, matches reference)
    const float* node_emb   = (const float*)d_in[4];
    const float* comp0  = (const float*)d_in[5];
    const float* bases0 = (const float*)d_in[6];
    const float* root0  = (const float*)d_in[7];
    const float* bias0  = (const float*)d_in[8];
    const float* comp1  = (const float*)d_in[9];
    const float* bases1 = (const float*)d_in[10];
    const float* root1  = (const float*)d_in[11];
    const float* bias1  = (const float*)d_in[12];
    float* out = (float*)d_out;

    // Workspace carve-up (~436 MB total)
    char* ws = (char*)d_ws;
    size_t off = 0;
    auto carve = [&](size_t bytes) {
        void* p = ws + off;
        off = (off + bytes + 255) & ~(size_t)255;
        return p;
    };
    const size_t S_bytes   = (size_t)NNODES * KREL * sizeof(float);
    const size_t cnt_bytes = (size_t)NNODES * RREL * sizeof(float);
    float*  S     = (float*)carve(S_bytes);
    float*  cnt   = (float*)carve(cnt_bytes);
    float2* Wpack = (float2*)carve((size_t)KTOT * DDIM * sizeof(float));
    float*  xg    = (float*)carve((size_t)NNODES * DDIM * sizeof(float));
    float*  h     = (float*)carve((size_t)NNODES * DDIM * sizeof(float));

    const int gather_blocks  = (NNODES * 32 + 255) / 256;
    const int scatter_blocks = (NEDGES * 32 + 255) / 256;
    const int invcnt_blocks  = (NNODES * RREL + 255) / 256;
    const int wcat_blocks    = (KSTEPS * 256 + 255) / 256;   // 1312
    const int gemm_blocks    = NNODES / 32;                  // 625, exact

    // x = node_emb[node_index]
    rgcn_gather_kernel<<<gather_blocks, 256, 0, stream>>>(node_index, node_emb, xg);

    // ---- layer 0 ----
    hipMemsetAsync(S, 0, S_bytes, stream);
    hipMemsetAsync(cnt, 0, cnt_bytes, stream);
    rgcn_scatter_kernel<<<scatter_blocks, 256, 0, stream>>>(edge_index, edge_type, xg, S, cnt);
    rgcn_invcnt_kernel<<<invcnt_blocks, 256, 0, stream>>>(cnt);
    rgcn_wcat_kernel<<<wcat_blocks, 256, 0, stream>>>(comp0, bases0, root0, Wpack);
    rgcn_gemm_kernel<<<gemm_blocks, 128, 0, stream>>>(S, cnt, xg, Wpack, bias0, h);

    // ---- layer 1 ----
    hipMemsetAsync(S, 0, S_bytes, stream);
    hipMemsetAsync(cnt, 0, cnt_bytes, stream);
    rgcn_scatter_kernel<<<scatter_blocks, 256, 0, stream>>>(edge_index, edge_type, h, S, cnt);
    rgcn_invcnt_kernel<<<invcnt_blocks, 256, 0, stream>>>(cnt);
    rgcn_wcat_kernel<<<wcat_blocks, 256, 0, stream>>>(comp1, bases1, root1, Wpack);
    rgcn_gemm_kernel<<<gemm_blocks, 128, 0, stream>>>(S, cnt, h, Wpack, bias1, out);
}